// Mean_Aggregator_37254546326089
// MI455X (gfx1250) — compile-verified
//
#include <hip/hip_runtime.h>

#define N_NODES 50000
#define N_EDGES 800000
#define N_FEAT  64
#define UNITS   128

typedef __attribute__((ext_vector_type(2))) float v2f;
typedef __attribute__((ext_vector_type(8))) float v8f;

// ---------------------------------------------------------------------------
// Phase 0: zero the segment-sum + count accumulators in workspace.
// ---------------------------------------------------------------------------
__global__ void zero_ws_kernel(float* __restrict__ ws, int n) {
    int i = blockIdx.x * blockDim.x + threadIdx.x;
    if (i < n) ws[i] = 0.0f;
}

// ---------------------------------------------------------------------------
// Phase 1: edge scatter. One thread per (edge, feature). Consecutive threads
// cover consecutive features of one edge -> 256B coalesced gather of x[col]
// and 256B-grouped atomics into seg[row]. Working set is L2-resident (192MB),
// so hardware f32 atomics (relaxed, agent scope -> global_atomic_add_f32)
// run at L2 atomic throughput, no CAS loops.
// ---------------------------------------------------------------------------
__global__ void scatter_edges_kernel(const float* __restrict__ x,
                                     const int*   __restrict__ row,   // targets
                                     const int*   __restrict__ col,   // sources
                                     float* __restrict__ seg,
                                     float* __restrict__ cnt) {
    long long gid = (long long)blockIdx.x * blockDim.x + threadIdx.x;
    if (gid >= (long long)N_EDGES * N_FEAT) return;
    int e = (int)(gid >> 6);    // gid / 64
    int f = (int)(gid & 63);    // gid % 64
    int r = row[e];
    int c = col[e];
    float v = x[(long long)c * N_FEAT + f];
    __hip_atomic_fetch_add(&seg[(long long)r * N_FEAT + f], v,
                           __ATOMIC_RELAXED, __HIP_MEMORY_SCOPE_AGENT);
    if (f == 0) {
        __hip_atomic_fetch_add(&cnt[r], 1.0f,
                               __ATOMIC_RELAXED, __HIP_MEMORY_SCOPE_AGENT);
    }
}

// ---------------------------------------------------------------------------
// Phase 2: fused mean + dual GEMM + bias + ReLU using V_WMMA_F32_16X16X4_F32.
// Block = 512 threads = 16 wave32 waves, one 16-row node slab per block
// (50000 % 16 == 0, so EXEC is all-ones everywhere -> WMMA legal).
// Waves 0..7  : neighbor GEMM, A = seg * (1/max(cnt,1)), W = neighs_kernel,
//               output columns [0,128).
// Waves 8..15 : self GEMM, A = x, W = self_kernel, output columns [128,256).
// Each wave owns one 16x16 output tile; K=64 -> 16 WMMA steps of K=4.
//
// f32 16x16x4 VGPR layouts (ISA 7.12.2):
//   A (16x4):  lane m = lane%16 is row M; K = k0 + 2*(lane/16) + {vgpr0,vgpr1}
//   B (4x16):  lane n = lane%16 is col N; K mirrored like A
//   C/D:       vgpr i -> row M = i + 8*(lane/16), col N = lane%16
// ---------------------------------------------------------------------------
__global__ void __launch_bounds__(512)
sage_wmma_gemm_kernel(const float* __restrict__ x,
                      const float* __restrict__ seg,
                      const float* __restrict__ cnt,
                      const float* __restrict__ Wn,
                      const float* __restrict__ Ws,
                      const float* __restrict__ bias,
                      float* __restrict__ out) {
    const int lane    = threadIdx.x & 31;
    const int wave    = threadIdx.x >> 5;      // 0..15
    const int rowbase = blockIdx.x * 16;

    const bool isSelf  = (wave >= 8);
    const int  colTile = (wave & 7) * 16;

    const int n = lane & 15;                   // A row / B col within tile
    const int g = lane >> 4;                   // 0 or 1 (K-half selector)

    const float* __restrict__ A = isSelf ? x  : seg;
    const float* __restrict__ W = isSelf ? Ws : Wn;

    const int m = rowbase + n;                 // global node row this lane reads
    const float inv = isSelf ? 1.0f : (1.0f / fmaxf(cnt[m], 1.0f));

    v8f c = {};                                // f32 accumulator, 8 VGPRs
    #pragma unroll
    for (int k0 = 0; k0 < N_FEAT; k0 += 4) {
        const int ka = k0 + 2 * g;
        v2f a, b;
        a.x = A[(long long)m * N_FEAT + ka    ] * inv;
        a.y = A[(long long)m * N_FEAT + ka + 1] * inv;
        b.x = W[(ka    ) * UNITS + colTile + n];
        b.y = W[(ka + 1) * UNITS + colTile + n];
        // 8 args: (neg_a, A, neg_b, B, c_mod, C, reuse_a, reuse_b)
        c = __builtin_amdgcn_wmma_f32_16x16x4_f32(
                false, a, false, b, (short)0, c, false, false);
    }

    const int   outcol = (isSelf ? UNITS : 0) + colTile + n;
    const float bv     = bias[outcol];
    #pragma unroll
    for (int i = 0; i < 8; ++i) {
        const int r = rowbase + i + 8 * g;     // C/D row mapping
        float v = c[i] + bv;
        out[(long long)r * (2 * UNITS) + outcol] = (v > 0.0f) ? v : 0.0f;
    }
}

// ---------------------------------------------------------------------------
// Launch wrapper.
// Inputs (setup_inputs order):
//   d_in[0] x            [50000, 64]  f32
//   d_in[1] edge_index   [2, 800000]  int32 (row = targets, col = sources)
//   d_in[2] neighs_kernel[64, 128]    f32
//   d_in[3] self_kernel  [64, 128]    f32
//   d_in[4] bias         [256]        f32
// Output: [50000, 256] f32
// Workspace: seg [50000*64] f32 + cnt [50000] f32  (~13 MB)
// ---------------------------------------------------------------------------
extern "C" void kernel_launch(void* const* d_in, const int* in_sizes, int n_in,
                              void* d_out, int out_size, void* d_ws, size_t ws_size,
                              hipStream_t stream) {
    const float* x    = (const float*)d_in[0];
    const int*   ei   = (const int*)  d_in[1];
    const float* Wn   = (const float*)d_in[2];
    const float* Ws   = (const float*)d_in[3];
    const float* bias = (const float*)d_in[4];
    float*       out  = (float*)d_out;

    float* seg = (float*)d_ws;
    float* cnt = seg + (size_t)N_NODES * N_FEAT;

    // Phase 0: zero accumulators
    const int nzero = N_NODES * N_FEAT + N_NODES;
    zero_ws_kernel<<<(nzero + 255) / 256, 256, 0, stream>>>(seg, nzero);

    // Phase 1: scatter-add (gather x[col], atomic add into seg[row], count)
    const long long work = (long long)N_EDGES * N_FEAT;
    scatter_edges_kernel<<<(int)((work + 255) / 256), 256, 0, stream>>>(
        x, ei /*row*/, ei + N_EDGES /*col*/, seg, cnt);

    // Phase 2: fused mean + dual WMMA GEMM + bias + ReLU
    sage_wmma_gemm_kernel<<<N_NODES / 16, 512, 0, stream>>>(
        x, seg, cnt, Wn, Ws, bias, out);
}